// sAttention_28690381537434
// MI455X (gfx1250) — compile-verified
//
#include <hip/hip_runtime.h>
#include <math.h>

#define H 1024
#define S 2048
#define B 16
#define R (B * S)  // 32768 rows of b, flattened [B*S, H]

typedef __attribute__((ext_vector_type(2))) float v2f;
typedef __attribute__((ext_vector_type(4))) float v4f;
typedef __attribute__((ext_vector_type(8))) float v8f;

// Only meaningful for the device pass; host clang doesn't declare AMDGCN builtins.
#if defined(__HIP_DEVICE_COMPILE__) && !__has_builtin(__builtin_amdgcn_wmma_f32_16x16x4_f32)
#error "device pass: missing __builtin_amdgcn_wmma_f32_16x16x4_f32"
#endif

// ---------------------------------------------------------------------------
// Phase A: va[i] = Wa[i,:]·wc, u[i] = Wb[i,:]·wc; block 0 also computes
//          s0 = (ba+bb)·wc + bc.
// ---------------------------------------------------------------------------
__global__ void proj_kernel(const float* __restrict__ Wa, const float* __restrict__ Wb,
                            const float* __restrict__ wc, const float* __restrict__ ba,
                            const float* __restrict__ bb, const float* __restrict__ bc,
                            float* __restrict__ va, float* __restrict__ u,
                            float* __restrict__ s0) {
  __shared__ float redA[256];
  __shared__ float redB[256];
  const int i = blockIdx.x;
  const int t = threadIdx.x;

  const v4f* ra = (const v4f*)(Wa + (size_t)i * H);
  const v4f* rb = (const v4f*)(Wb + (size_t)i * H);
  const v4f* w4 = (const v4f*)wc;

  v4f xa = ra[t];
  v4f xb = rb[t];
  v4f ww = w4[t];
  v4f pa = xa * ww;
  v4f pb = xb * ww;
  redA[t] = pa.x + pa.y + pa.z + pa.w;
  redB[t] = pb.x + pb.y + pb.z + pb.w;
  __syncthreads();
  for (int o = 128; o > 0; o >>= 1) {
    if (t < o) {
      redA[t] += redA[t + o];
      redB[t] += redB[t + o];
    }
    __syncthreads();
  }
  if (t == 0) {
    va[i] = redA[0];
    u[i] = redB[0];
  }

  if (i == 0) {
    __syncthreads();
    const v4f* b4a = (const v4f*)ba;
    const v4f* b4b = (const v4f*)bb;
    v4f sum4 = (b4a[t] + b4b[t]) * ww;
    redA[t] = sum4.x + sum4.y + sum4.z + sum4.w;
    __syncthreads();
    for (int o = 128; o > 0; o >>= 1) {
      if (t < o) redA[t] += redA[t + o];
      __syncthreads();
    }
    if (t == 0) s0[0] = redA[0] + bc[0];
  }
}

// ---------------------------------------------------------------------------
// Phase 1: ca[s] = a[s,:]·va
// ---------------------------------------------------------------------------
__global__ void ca_kernel(const float* __restrict__ a, const float* __restrict__ va,
                          float* __restrict__ ca) {
  __shared__ float red[256];
  const int s = blockIdx.x;
  const int t = threadIdx.x;
  v4f x = ((const v4f*)(a + (size_t)s * H))[t];
  v4f v = ((const v4f*)va)[t];
  v4f p = x * v;
  red[t] = p.x + p.y + p.z + p.w;
  __syncthreads();
  for (int o = 128; o > 0; o >>= 1) {
    if (t < o) red[t] += red[t + o];
    __syncthreads();
  }
  if (t == 0) ca[s] = red[0];
}

// ---------------------------------------------------------------------------
// Phase 2 (WMMA): raw[r] = b_flat[r,:]·u  for 16 rows per wave, via
// V_WMMA_F32_16X16X4_F32.  A = 16x4 tile of b rows (documented layout:
// VGPR0 holds K=k0 (lanes 0-15) / K=k0+2 (lanes 16-31), VGPR1 the K+1s,
// row M = lane&15).  B = 4x16 broadcast of u[k0..k0+3] into every column
// (same per-lane K pattern), so D[m,n] accumulates the row-dot for all n.
// u staged in LDS.
// ---------------------------------------------------------------------------
__global__ void score_wmma_kernel(const float* __restrict__ bmat,
                                  const float* __restrict__ u,
                                  float* __restrict__ raw) {
  __shared__ v4f ulds4[H / 4];
  const int t = threadIdx.x;
  ulds4[t] = ((const v4f*)u)[t];
  __syncthreads();
  const float* ulds = (const float*)ulds4;

  const int wave = t >> 5;
  const int lane = t & 31;
  const int half = lane >> 4;   // 0: K={k0,k0+1}, 1: K={k0+2,k0+3}
  const int m = lane & 15;      // row within the 16-row tile
  const int rowBase = blockIdx.x * 128 + wave * 16;

  const float* arow = bmat + (size_t)(rowBase + m) * H + 2 * half;
  const float* up = ulds + 2 * half;

  v8f acc = {};
#pragma unroll 8
  for (int k0 = 0; k0 < H; k0 += 4) {
    v2f av = *(const v2f*)(arow + k0);  // b rows -> A fragment
    v2f bv = *(const v2f*)(up + k0);    // u broadcast -> B fragment
    acc = __builtin_amdgcn_wmma_f32_16x16x4_f32(
        /*neg_a=*/false, av, /*neg_b=*/false, bv,
        /*c_mod=*/(short)0, acc, /*reuse_a=*/false, /*reuse_b=*/false);
  }

  // D layout: VGPR j, lanes 0-15 => row M=j, lanes 16-31 => row M=j+8.
  // Every column n holds the same value; column n==0 (lanes 0 and 16) writes.
  if (m == 0) {
    float* dst = raw + rowBase + 8 * half;
    v4f lo = {acc[0], acc[1], acc[2], acc[3]};
    v4f hi = {acc[4], acc[5], acc[6], acc[7]};
    *(v4f*)(dst + 0) = lo;
    *(v4f*)(dst + 4) = hi;
  }
}

// ---------------------------------------------------------------------------
// Phase 3: softmax over S per batch b; scores = raw + ca + s0
// ---------------------------------------------------------------------------
__global__ void softmax_kernel(const float* __restrict__ raw, const float* __restrict__ ca,
                               const float* __restrict__ s0p, float* __restrict__ att) {
  __shared__ float red[256];
  const int b = blockIdx.x;
  const int t = threadIdx.x;
  const float s0 = s0p[0];

  float v[8];
  float mx = -1e30f;
#pragma unroll
  for (int i = 0; i < 8; ++i) {
    int s = t + i * 256;
    v[i] = raw[b * S + s] + ca[s] + s0;
    mx = fmaxf(mx, v[i]);
  }
  red[t] = mx;
  __syncthreads();
  for (int o = 128; o > 0; o >>= 1) {
    if (t < o) red[t] = fmaxf(red[t], red[t + o]);
    __syncthreads();
  }
  const float M = red[0];
  __syncthreads();

  float sum = 0.f;
#pragma unroll
  for (int i = 0; i < 8; ++i) {
    v[i] = expf(v[i] - M);
    sum += v[i];
  }
  red[t] = sum;
  __syncthreads();
  for (int o = 128; o > 0; o >>= 1) {
    if (t < o) red[t] += red[t + o];
    __syncthreads();
  }
  const float inv = 1.0f / red[0];
#pragma unroll
  for (int i = 0; i < 8; ++i) att[b * S + t + i * 256] = v[i] * inv;
}

// ---------------------------------------------------------------------------
// Phase 4: out[s,h] = sum_b att[b,s] * b[b,s,h]   (second pass over b hits L2)
// ---------------------------------------------------------------------------
__global__ void out_kernel(const float* __restrict__ bmat, const float* __restrict__ att,
                           float* __restrict__ out) {
  __shared__ float w[B];
  const int s = blockIdx.x;
  const int t = threadIdx.x;
  if (t < B) w[t] = att[t * S + s];
  __syncthreads();

  v4f acc = {0.f, 0.f, 0.f, 0.f};
#pragma unroll
  for (int bb = 0; bb < B; ++bb) {
    const v4f* row = (const v4f*)(bmat + ((size_t)bb * S + (size_t)s) * H);
    acc = acc + row[t] * w[bb];
  }
  ((v4f*)(out + (size_t)s * H))[t] = acc;
}

// ---------------------------------------------------------------------------
extern "C" void kernel_launch(void* const* d_in, const int* in_sizes, int n_in,
                              void* d_out, int out_size, void* d_ws, size_t ws_size,
                              hipStream_t stream) {
  const float* a = (const float*)d_in[0];   // [S, H]
  const float* bm = (const float*)d_in[1];  // [B, S, H]
  const float* Wa = (const float*)d_in[2];  // [H, H]
  const float* ba = (const float*)d_in[3];  // [H]
  const float* Wb = (const float*)d_in[4];  // [H, H]
  const float* bb = (const float*)d_in[5];  // [H]
  const float* wc = (const float*)d_in[6];  // [H]
  const float* bc = (const float*)d_in[7];  // scalar
  float* out = (float*)d_out;               // [S, H]

  float* ws = (float*)d_ws;
  float* va = ws;               // H
  float* u = ws + 1024;         // H
  float* ca = ws + 2048;        // S
  float* s0 = ws + 4096;        // 1 (padded to 64)
  float* raw = ws + 4160;       // B*S
  float* att = ws + 4160 + R;   // B*S
  // total: ~279 KB of workspace

  proj_kernel<<<H, 256, 0, stream>>>(Wa, Wb, wc, ba, bb, bc, va, u, s0);
  ca_kernel<<<S, 256, 0, stream>>>(a, va, ca);
  score_wmma_kernel<<<R / 128, 256, 0, stream>>>(bm, u, raw);
  softmax_kernel<<<B, 256, 0, stream>>>(raw, ca, s0, att);
  out_kernel<<<S, 256, 0, stream>>>(bm, att, out);
}